// GCNContext_20426864460532
// MI455X (gfx1250) — compile-verified
//
#include <hip/hip_runtime.h>

#define NN  50000
#define EE  400000
#define BB  256
#define SS  50
#define CTX 50
#define EMB 200
#define HID 256
#define GHD 100
#define G3  300
#define KMAX 400            // largest GEMM K (fc1)

typedef __attribute__((ext_vector_type(2))) float v2f;
typedef __attribute__((ext_vector_type(8))) float v8f;

#if defined(__gfx1250__) && __has_builtin(__builtin_amdgcn_global_load_async_to_lds_b32)
#define HAS_ASYNC_LDS 1
#else
#define HAS_ASYNC_LDS 0
#endif

typedef __attribute__((address_space(1))) int  gint_t;
typedef __attribute__((address_space(3))) int  lint_t;

__device__ __forceinline__ void wait_async_zero() {
#if defined(__gfx1250__)
#if __has_builtin(__builtin_amdgcn_s_wait_asynccnt)
    __builtin_amdgcn_s_wait_asynccnt(0);
#else
    asm volatile("s_wait_asynccnt 0x0" ::: "memory");
#endif
#endif
}

// ---------------------------------------------------------------------------
// WMMA f32 GEMM: D[M,N] = act(A[M,K] * op(B) + bias), op(B)=B (TRANSB=0) or B^T.
// Block = 128 threads (4 waves). Block computes a 64x16 output slab:
//   - B panel (K x 16) staged once into LDS (async global->LDS when full tile)
//   - each wave owns one 16x16 tile, V_WMMA_F32_16X16X4_F32 over K/4 steps
// Fragment layouts per CDNA5 ISA 7.12.2:
//   A 16x4 : lanes 0-15 hold K={k0,k0+1}; lanes 16-31 hold K={k0+2,k0+3}
//   B 4x16 : same K split; N = lane&15
//   C/D    : vgpr i, lane l -> row = i + 8*(l>=16), col = l&15
// ---------------------------------------------------------------------------
template <int TRANSB>
__global__ void wmma_gemm_f32(const float* __restrict__ A, int lda,
                              const float* __restrict__ Bm, int ldb,
                              const float* __restrict__ bias,
                              float* __restrict__ D, int ldd,
                              int M, int N, int K, int relu)
{
    __shared__ float Bs[KMAX * 16];

    const int tid  = threadIdx.x;           // 0..127
    const int wave = tid >> 5;
    const int lane = tid & 31;
    const int half = lane >> 4;             // 0 or 1
    const int l16  = lane & 15;

    const int tileN = blockIdx.y * 16;
    const int tileM = blockIdx.x * 64 + wave * 16;

    // ---- stage B panel (K x 16, row-major k-major) into LDS ----
    if (tileN + 16 <= N) {                  // block-uniform: full tile
        for (int idx = tid; idx < K * 16; idx += 128) {
            const int k = idx >> 4, n = idx & 15;
            const float* gp = TRANSB ? &Bm[(long)(tileN + n) * ldb + k]
                                     : &Bm[(long)k * ldb + tileN + n];
#if HAS_ASYNC_LDS
            __builtin_amdgcn_global_load_async_to_lds_b32(
                (gint_t*)gp, (lint_t*)&Bs[idx], 0, 0);
#else
            Bs[idx] = *gp;
#endif
        }
#if HAS_ASYNC_LDS
        wait_async_zero();
#endif
    } else {                                // ragged N edge: zero-padded scalar path
        for (int idx = tid; idx < K * 16; idx += 128) {
            const int k = idx >> 4, n = idx & 15;
            float v = 0.0f;
            if (tileN + n < N)
                v = TRANSB ? Bm[(long)(tileN + n) * ldb + k]
                           : Bm[(long)k * ldb + tileN + n];
            Bs[idx] = v;
        }
    }
    __syncthreads();

    // ---- accumulate: one 16x16 tile per wave ----
    const int amr = tileM + l16;
    const int am  = (amr < M) ? amr : (M - 1);   // clamped load row (value-safe)
    v8f c = {};
    for (int k0 = 0; k0 < K; k0 += 4) {
        const int ka = k0 + 2 * half;
        v2f a, b;
        a.x = A[(long)am * lda + ka];
        a.y = A[(long)am * lda + ka + 1];
        b.x = Bs[ka * 16 + l16];
        b.y = Bs[(ka + 1) * 16 + l16];
        c = __builtin_amdgcn_wmma_f32_16x16x4_f32(
                false, a, false, b, (short)0, c, false, false);
    }

    // ---- epilogue: bias + act + masked store ----
    const int bn  = tileN + l16;
    const int bnc = (bn < N) ? bn : (N - 1);
    const float bv = bias ? bias[bnc] : 0.0f;
    const bool bnOK = (bn < N);
#pragma unroll
    for (int i = 0; i < 8; ++i) {
        const int m = tileM + i + 8 * half;
        if (bnOK && m < M) {
            float v = c[i] + bv;
            if (relu) v = fmaxf(v, 0.0f);
            D[(long)m * ldd + bn] = v;
        }
    }
}

// ---------------------------------------------------------------------------
__global__ void fill_kernel(float* __restrict__ p, long n, float v) {
    long i = (long)blockIdx.x * blockDim.x + threadIdx.x;
    if (i < n) p[i] = v;
}

__global__ void scatter_feat(const float* __restrict__ lin, float* __restrict__ agg,
                             const int* __restrict__ src, const int* __restrict__ dst,
                             int nfeat, long total) {
    long i = (long)blockIdx.x * blockDim.x + threadIdx.x;
    if (i >= total) return;
    int e = (int)(i / nfeat);
    int f = (int)(i % nfeat);
    int s = src[e], d = dst[e];
    atomicAdd(&agg[(long)d * nfeat + f], lin[(long)s * nfeat + f]);
}

__global__ void deg_kernel(const int* __restrict__ dst, float* __restrict__ deg, int e) {
    int i = blockIdx.x * blockDim.x + threadIdx.x;
    if (i < e) atomicAdd(&deg[dst[i]], 1.0f);
}

__global__ void norm_kernel(float* __restrict__ buf, const float* __restrict__ deg,
                            int nfeat, long total, int relu) {
    long i = (long)blockIdx.x * blockDim.x + threadIdx.x;
    if (i >= total) return;
    int row = (int)(i / nfeat);
    float c = deg[row];
    c = (c < 1.0f) ? 1.0f : c;
    float v = buf[i] / c;
    if (relu) v = fmaxf(v, 0.0f);
    buf[i] = v;
}

__global__ void gather_sent(const float* __restrict__ h, const int* __restrict__ idx,
                            float* __restrict__ out, long total) {
    long i = (long)blockIdx.x * blockDim.x + threadIdx.x;
    if (i >= total) return;
    int bs = (int)(i / EMB);
    int f  = (int)(i % EMB);
    int node = idx[bs];
    out[i] = h[(long)node * EMB + f];
}

__global__ void gru_gate(const float* __restrict__ gi, int giStride,
                         const float* __restrict__ gh,
                         const float* __restrict__ hold,
                         float* __restrict__ hnew) {
    int i = blockIdx.x * blockDim.x + threadIdx.x;
    if (i >= BB * GHD) return;
    int b = i / GHD, j = i % GHD;
    const float* gib = gi + (long)b * giStride;
    const float* ghb = gh + (long)b * G3;
    float ir = gib[j], iz = gib[GHD + j], in_ = gib[2 * GHD + j];
    float hr = ghb[j], hz = ghb[GHD + j], hn  = ghb[2 * GHD + j];
    float r = 1.0f / (1.0f + __expf(-(ir + hr)));
    float z = 1.0f / (1.0f + __expf(-(iz + hz)));
    float n = tanhf(in_ + r * hn);
    hnew[i] = (1.0f - z) * n + z * hold[i];
}

__global__ void cont_kernel(const float* __restrict__ h, const int* __restrict__ context,
                            float* __restrict__ out, int total) {
    int i = blockIdx.x * blockDim.x + threadIdx.x;
    if (i >= total) return;
    int b = i / EMB, f = i % EMB;
    float acc = 0.0f;
    for (int c = 0; c < CTX; ++c) {
        int node = context[b * CTX + c];
        if (c + 1 < CTX)
            __builtin_prefetch(&h[(long)context[b * CTX + c + 1] * EMB + f], 0, 1);
        acc += h[(long)node * EMB + f];
    }
    out[i] = acc;
}

__global__ void concat_kernel(const float* __restrict__ hf, const float* __restrict__ hb,
                              const float* __restrict__ cont, float* __restrict__ emd,
                              int total) {
    int i = blockIdx.x * blockDim.x + threadIdx.x;
    if (i >= total) return;
    int b = i / (2 * EMB), j = i % (2 * EMB);
    float v;
    if (j < GHD)            v = hf[b * GHD + j];
    else if (j < 2 * GHD)   v = hb[b * GHD + (j - GHD)];
    else                    v = cont[b * EMB + (j - 2 * GHD)];
    emd[i] = v;
}

__global__ void fc2_kernel(const float* __restrict__ in, const float* __restrict__ w,
                           const float* __restrict__ b, float* __restrict__ out, int total) {
    int i = blockIdx.x * blockDim.x + threadIdx.x;
    if (i >= total) return;
    int row = i / 2, o = i % 2;
    float acc = b[o];
    for (int k = 0; k < 512; ++k)
        acc += in[row * 512 + k] * w[o * 512 + k];
    out[i] = acc;
}

// ---------------------------------------------------------------------------
extern "C" void kernel_launch(void* const* d_in, const int* in_sizes, int n_in,
                              void* d_out, int out_size, void* d_ws, size_t ws_size,
                              hipStream_t stream) {
    const float* x        = (const float*)d_in[0];
    const int*   edge     = (const int*)d_in[1];
    const int*   sentence = (const int*)d_in[2];
    const int*   context  = (const int*)d_in[3];
    const float* conv1_w  = (const float*)d_in[4];
    const float* conv1_b  = (const float*)d_in[5];
    const float* conv2_w  = (const float*)d_in[6];
    const float* conv2_b  = (const float*)d_in[7];
    const float* wih_f    = (const float*)d_in[8];
    const float* whh_f    = (const float*)d_in[9];
    const float* bih_f    = (const float*)d_in[10];
    const float* bhh_f    = (const float*)d_in[11];
    const float* wih_b    = (const float*)d_in[12];
    const float* whh_b    = (const float*)d_in[13];
    const float* bih_b    = (const float*)d_in[14];
    const float* bhh_b    = (const float*)d_in[15];
    const float* fc1_w    = (const float*)d_in[16];
    const float* fc1_b    = (const float*)d_in[17];
    const float* fc2_w    = (const float*)d_in[18];
    const float* fc2_b    = (const float*)d_in[19];

    float* ws   = (float*)d_ws;
    float* bufA = ws;                 // N*HID = 12.8M floats
    float* bufB = ws + 12800000;      // N*HID
    float* bufC = ws + 25600000;      // N*EMB = 10M floats
    float* deg  = ws + 35600000;      // N

    const int* src = edge;            // edge_index[0]
    const int* dst = edge + EE;       // edge_index[1]

    // ---- conv1 linear: bufA = x @ conv1_w + b1  [N,HID] ----
    wmma_gemm_f32<0><<<dim3((NN + 63) / 64, HID / 16), 128, 0, stream>>>(
        x, EMB, conv1_w, HID, conv1_b, bufA, HID, NN, HID, EMB, 0);

    // ---- mean scatter + relu -> bufB ----
    fill_kernel<<<(NN * HID + 255) / 256, 256, 0, stream>>>(bufB, (long)NN * HID, 0.0f);
    fill_kernel<<<(NN + 255) / 256, 256, 0, stream>>>(deg, (long)NN, 0.0f);
    long tot1 = (long)EE * HID;
    scatter_feat<<<(int)((tot1 + 255) / 256), 256, 0, stream>>>(bufA, bufB, src, dst, HID, tot1);
    deg_kernel<<<(EE + 255) / 256, 256, 0, stream>>>(dst, deg, EE);
    norm_kernel<<<(NN * HID + 255) / 256, 256, 0, stream>>>(bufB, deg, HID, (long)NN * HID, 1);

    // ---- conv2 linear: bufC = bufB @ conv2_w + b2  [N,EMB] ----
    wmma_gemm_f32<0><<<dim3((NN + 63) / 64, (EMB + 15) / 16), 128, 0, stream>>>(
        bufB, HID, conv2_w, EMB, conv2_b, bufC, EMB, NN, EMB, HID, 0);

    // ---- mean scatter (no relu) -> h = bufA [N,EMB] ----
    fill_kernel<<<(NN * EMB + 255) / 256, 256, 0, stream>>>(bufA, (long)NN * EMB, 0.0f);
    long tot2 = (long)EE * EMB;
    scatter_feat<<<(int)((tot2 + 255) / 256), 256, 0, stream>>>(bufC, bufA, src, dst, EMB, tot2);
    norm_kernel<<<(NN * EMB + 255) / 256, 256, 0, stream>>>(bufA, deg, EMB, (long)NN * EMB, 0);
    float* h = bufA;

    // ---- GRU-phase sub-buffers (bufB / bufC regions now free) ----
    float* sent  = bufB;                // [B,S,EMB] 2,560,000
    float* gh    = bufB + 2560000;      // [B,3H]       76,800
    float* hA    = bufB + 2640000;      // [B,H]        25,600
    float* hB    = bufB + 2665600;
    float* hz    = bufB + 2691200;      // permanent zeros
    float* hbout = bufB + 2716800;
    float* gib   = bufB + 2742400;      // [B,3H]
    float* cont  = bufB + 2819200;      // [B,EMB]
    float* emd   = bufB + 2870400;      // [B,2*EMB]
    float* fc1o  = bufB + 2972800;      // [B,512]
    float* gif   = bufC;                // [B*S,3H] 3,840,000

    gather_sent<<<(BB * SS * EMB + 255) / 256, 256, 0, stream>>>(
        h, sentence, sent, (long)BB * SS * EMB);

    // input-side GRU GEMM hoisted out of the time loop: gif = sent @ wih_f^T + bih_f
    wmma_gemm_f32<1><<<dim3((BB * SS + 63) / 64, (G3 + 15) / 16), 128, 0, stream>>>(
        sent, EMB, wih_f, EMB, bih_f, gif, G3, BB * SS, G3, EMB, 0);

    fill_kernel<<<(BB * GHD + 255) / 256, 256, 0, stream>>>(hA, (long)BB * GHD, 0.0f);
    fill_kernel<<<(BB * GHD + 255) / 256, 256, 0, stream>>>(hB, (long)BB * GHD, 0.0f);
    fill_kernel<<<(BB * GHD + 255) / 256, 256, 0, stream>>>(hz, (long)BB * GHD, 0.0f);

    // forward GRU: 50 sequential steps; only the 256x300x100 hidden GEMM recurs
    for (int t = 0; t < SS; ++t) {
        const float* hprev = (t & 1) ? hB : hA;
        float*       hnext = (t & 1) ? hA : hB;
        wmma_gemm_f32<1><<<dim3((BB + 63) / 64, (G3 + 15) / 16), 128, 0, stream>>>(
            hprev, GHD, whh_f, GHD, bhh_f, gh, G3, BB, G3, GHD, 0);
        gru_gate<<<(BB * GHD + 255) / 256, 256, 0, stream>>>(
            gif + (long)t * G3, SS * G3, gh, hprev, hnext);
    }
    const float* hf_last = hA;   // step 49 (odd) writes hA

    // backward GRU: out_b[0] is ONE step on x_{S-1} from h0=0
    wmma_gemm_f32<1><<<dim3((BB + 63) / 64, (G3 + 15) / 16), 128, 0, stream>>>(
        sent + (SS - 1) * EMB, SS * EMB, wih_b, EMB, bih_b, gib, G3, BB, G3, EMB, 0);
    wmma_gemm_f32<1><<<dim3((BB + 63) / 64, (G3 + 15) / 16), 128, 0, stream>>>(
        hz, GHD, whh_b, GHD, bhh_b, gh, G3, BB, G3, GHD, 0);
    gru_gate<<<(BB * GHD + 255) / 256, 256, 0, stream>>>(gib, G3, gh, hz, hbout);

    // context sum, concat, FC head
    cont_kernel<<<(BB * EMB + 255) / 256, 256, 0, stream>>>(h, context, cont, BB * EMB);
    concat_kernel<<<(BB * 2 * EMB + 255) / 256, 256, 0, stream>>>(
        hf_last, hbout, cont, emd, BB * 2 * EMB);
    wmma_gemm_f32<1><<<dim3((BB + 63) / 64, 512 / 16), 128, 0, stream>>>(
        emd, 2 * EMB, fc1_w, 2 * EMB, fc1_b, fc1o, 512, BB, 512, 2 * EMB, 1);
    fc2_kernel<<<(BB * 2 + 255) / 256, 256, 0, stream>>>(
        fc1o, fc2_w, fc2_b, (float*)d_out, BB * 2);
}